// PAALossComputation_83373905150006
// MI455X (gfx1250) — compile-verified
//
#include <hip/hip_runtime.h>
#include <math.h>

#define A_NUM 21824
#define B_NUM 32
#define G_NUM 32
#define C_NUM 80
#define IOU_THR_F 0.1f
#define ALPHA_F 0.25f
#define REG_W_F 1.3f
#define IOU_W_F 0.5f
#define EPS_F 1e-7f
#define DWH_CLAMP_F 4.135166556742356f /* log(1000/16) */

typedef __attribute__((ext_vector_type(2))) float v2f;
typedef __attribute__((ext_vector_type(4))) float v4f;
typedef __attribute__((ext_vector_type(8))) float v8f;

// ---------- shared math helpers (identical codegen between kernels matters for
// the iou == gt_best float-equality test, so both kernels inline this one fn) --
__device__ __forceinline__ float iou_one(float gx1, float gy1, float gx2, float gy2,
                                         float ax1, float ay1, float ax2, float ay2) {
    float area_g = (gx2 - gx1 + 1.0f) * (gy2 - gy1 + 1.0f);
    float area_a = (ax2 - ax1 + 1.0f) * (ay2 - ay1 + 1.0f);
    float ltx = fmaxf(gx1, ax1), lty = fmaxf(gy1, ay1);
    float rbx = fminf(gx2, ax2), rby = fminf(gy2, ay2);
    float w = fmaxf(rbx - ltx + 1.0f, 0.0f);
    float h = fmaxf(rby - lty + 1.0f, 0.0f);
    float inter = w * h;
    return inter / (area_g + area_a - inter);
}

__device__ __forceinline__ void decode_box(float d0, float d1, float d2, float d3,
                                           float aw, float ah, float acx, float acy,
                                           float& x1, float& y1, float& x2, float& y2) {
    float dx = d0 / 10.0f, dy = d1 / 10.0f;
    float dw = fminf(d2 / 5.0f, DWH_CLAMP_F);
    float dh = fminf(d3 / 5.0f, DWH_CLAMP_F);
    float pcx = dx * aw + acx, pcy = dy * ah + acy;
    float pw = expf(dw) * aw, ph = expf(dh) * ah;
    x1 = pcx - 0.5f * (pw - 1.0f);
    y1 = pcy - 0.5f * (ph - 1.0f);
    x2 = pcx + 0.5f * (pw - 1.0f);
    y2 = pcy + 0.5f * (ph - 1.0f);
}

// ---------- fused block-wide sum of FOUR floats per thread (256 threads) -----
// V_WMMA_F32_16X16X4_F32 with an all-ones A matrix: D[m][n] = sum_k B[k][n],
// so each WMMA folds 64 LDS values into 16 column sums at full f32 precision.
// One barrier, then four independent 4-deep WMMA accumulate chains (they can
// overlap in the matrix pipe), then 4 shfl_xor steps finish 16 -> 1 each.
__device__ __forceinline__ void block_sum4_256(float v0, float v1, float v2, float v3,
                                               float (*lds)[256],
                                               float& r0, float& r1, float& r2, float& r3) {
    const int tid = threadIdx.x;
    lds[0][tid] = v0;
    lds[1][tid] = v1;
    lds[2][tid] = v2;
    lds[3][tid] = v3;
    __syncthreads();
#if defined(__gfx1250__) && __has_builtin(__builtin_amdgcn_wmma_f32_16x16x4_f32)
    {
        const int lane = tid & 31;
        v2f ones;
        ones[0] = 1.0f;
        ones[1] = 1.0f;
        v8f a0 = {}, a1 = {}, a2 = {}, a3 = {};
#pragma unroll
        for (int j = 0; j < 4; ++j) {
            const int i0 = lane + 64 * j;
            const int i1 = lane + 32 + 64 * j;
            v2f b0, b1, b2, b3;
            b0[0] = lds[0][i0]; b0[1] = lds[0][i1];
            b1[0] = lds[1][i0]; b1[1] = lds[1][i1];
            b2[0] = lds[2][i0]; b2[1] = lds[2][i1];
            b3[0] = lds[3][i0]; b3[1] = lds[3][i1];
            a0 = __builtin_amdgcn_wmma_f32_16x16x4_f32(false, ones, false, b0, (short)0, a0, false, false);
            a1 = __builtin_amdgcn_wmma_f32_16x16x4_f32(false, ones, false, b1, (short)0, a1, false, false);
            a2 = __builtin_amdgcn_wmma_f32_16x16x4_f32(false, ones, false, b2, (short)0, a2, false, false);
            a3 = __builtin_amdgcn_wmma_f32_16x16x4_f32(false, ones, false, b3, (short)0, a3, false, false);
        }
        // aX[0] lane L = sum of all lds[X][i] with (i & 15) == (L & 15)
        float s0 = a0[0], s1 = a1[0], s2 = a2[0], s3 = a3[0];
#pragma unroll
        for (int m = 1; m <= 8; m <<= 1) {
            s0 += __shfl_xor(s0, m);
            s1 += __shfl_xor(s1, m);
            s2 += __shfl_xor(s2, m);
            s3 += __shfl_xor(s3, m);
        }
        r0 = s0; r1 = s1; r2 = s2; r3 = s3;
    }
#else
    {
        float s0 = 0.0f, s1 = 0.0f, s2 = 0.0f, s3 = 0.0f;
        for (int i = 0; i < 256; ++i) {
            s0 += lds[0][i];
            s1 += lds[1][i];
            s2 += lds[2][i];
            s3 += lds[3][i];
        }
        r0 = s0; r1 = s1; r2 = s2; r3 = s3;
    }
#endif
    __syncthreads();
}

// ---------- kernel 1: per-GT best IoU over all anchors -----------------------
__global__ void paa_gtbest_kernel(const float* __restrict__ anchors,
                                  const float* __restrict__ gt_boxes,
                                  float* __restrict__ ws_gtbest) {
    __shared__ float4 s_gt[G_NUM];
    __shared__ int s_best[G_NUM];
    const int b = blockIdx.x;
    const int tid = threadIdx.x;
    if (tid < G_NUM) {
        s_gt[tid] = reinterpret_cast<const float4*>(gt_boxes + (size_t)b * G_NUM * 4)[tid];
        s_best[tid] = 0; // iou >= 0, so int-bit max over 0-init is exact
    }
    __syncthreads();
    float best[G_NUM];
#pragma unroll
    for (int g = 0; g < G_NUM; ++g) best[g] = 0.0f;
    for (int a = tid; a < A_NUM; a += blockDim.x) {
        float4 an = reinterpret_cast<const float4*>(anchors)[a];
#pragma unroll
        for (int g = 0; g < G_NUM; ++g) {
            float4 gb = s_gt[g];
            float i = iou_one(gb.x, gb.y, gb.z, gb.w, an.x, an.y, an.z, an.w);
            best[g] = fmaxf(best[g], i);
        }
    }
#pragma unroll
    for (int g = 0; g < G_NUM; ++g) atomicMax(&s_best[g], __float_as_int(best[g]));
    __syncthreads();
    if (tid < G_NUM) ws_gtbest[b * G_NUM + tid] = __int_as_float(s_best[tid]);
}

// ---------- kernel 2: fused matching + focal + GIoU + BCE streaming pass -----
__global__ void paa_main_kernel(const float* __restrict__ box_cls,
                                const float* __restrict__ box_reg,
                                const float* __restrict__ iou_pred,
                                const float* __restrict__ anchors,
                                const float* __restrict__ gt_boxes,
                                const int* __restrict__ gt_labels,
                                const float* __restrict__ ws_gtbest,
                                float* __restrict__ ws_part) {
    __shared__ float4 s_gt[G_NUM];
    __shared__ int s_lab[G_NUM];
    __shared__ float s_gbest[G_NUM];
    __shared__ float s_red[4][256];
    const int b = blockIdx.y;
    const int tid = threadIdx.x;
    const int a = blockIdx.x * blockDim.x + tid;
    if (tid < G_NUM) {
        s_gt[tid] = reinterpret_cast<const float4*>(gt_boxes + (size_t)b * G_NUM * 4)[tid];
        s_lab[tid] = gt_labels[b * G_NUM + tid];
        s_gbest[tid] = ws_gtbest[b * G_NUM + tid];
    }
    __syncthreads();

    float cls_p = 0.0f, reg_p = 0.0f, bce_p = 0.0f, pos_p = 0.0f;
    if (a < A_NUM) {
        float4 an = reinterpret_cast<const float4*>(anchors)[a];
        // ---- matching (argmax keeps first tie, like jnp.argmax) ----
        float bestv = -1.0f;
        int besti = 0;
        bool lowq = false;
#pragma unroll
        for (int g = 0; g < G_NUM; ++g) {
            float4 gb = s_gt[g];
            float i = iou_one(gb.x, gb.y, gb.z, gb.w, an.x, an.y, an.z, an.w);
            if (i > bestv) { bestv = i; besti = g; }
            lowq = lowq || (i == s_gbest[g]);
        }
        const int label = ((bestv >= IOU_THR_F) || lowq) ? s_lab[besti] : 0;

        // ---- sigmoid focal loss over 80 channels ----
        // box_cls is a 223 MB single-use stream (> 192 MB L2): use NT loads so
        // it bypasses cache residency instead of evicting reusable data.
        const size_t row = ((size_t)b * A_NUM + a) * C_NUM;
        const v4f* cp = reinterpret_cast<const v4f*>(box_cls + row);
        float cls_sum = 0.0f;
#pragma unroll 4
        for (int q = 0; q < C_NUM / 4; ++q) {
            v4f xv = __builtin_nontemporal_load(cp + q);
            float xs[4] = {xv.x, xv.y, xv.z, xv.w};
#pragma unroll
            for (int k = 0; k < 4; ++k) {
                const int c = q * 4 + k;
                float x = xs[k];
                float e = expf(-fabsf(x));      // one transcendental pair per channel
                float L = log1pf(e);
                float p = (x >= 0.0f) ? 1.0f / (1.0f + e) : e / (1.0f + e);
                float ls_pos = fminf(x, 0.0f) - L;   // log_sigmoid(x)
                float ls_neg = fminf(-x, 0.0f) - L;  // log_sigmoid(-x)
                float tpos = ALPHA_F * (1.0f - p) * (1.0f - p) * ls_pos;
                float tneg = (1.0f - ALPHA_F) * p * p * ls_neg;
                cls_sum += (label == c + 1) ? tpos : tneg;
            }
        }
        cls_p = -cls_sum;

        // ---- GIoU + IoU-pred BCE on positives ----
        if (label > 0) {
            pos_p = 1.0f;
            float aw = an.z - an.x + 1.0f, ah = an.w - an.y + 1.0f;
            float acx = an.x + 0.5f * aw, acy = an.y + 0.5f * ah;
            float4 gbx = s_gt[besti];
            float gw = gbx.z - gbx.x + 1.0f, gh = gbx.w - gbx.y + 1.0f;
            float gcx = gbx.x + 0.5f * gw, gcy = gbx.y + 0.5f * gh;
            // encode(matched_gt, anchor)
            float t0 = 10.0f * (gcx - acx) / aw;
            float t1 = 10.0f * (gcy - acy) / ah;
            float t2 = 5.0f * logf(gw / aw);
            float t3 = 5.0f * logf(gh / ah);
            // predicted deltas (single-use stream -> NT)
            const size_t ro = ((size_t)b * A_NUM + a) * 4;
            v4f pd = __builtin_nontemporal_load(
                reinterpret_cast<const v4f*>(box_reg + ro));
            float px1, py1, px2r, py2r;
            decode_box(pd.x, pd.y, pd.z, pd.w, aw, ah, acx, acy, px1, py1, px2r, py2r);
            float px2 = fmaxf(px1, px2r), py2 = fmaxf(py1, py2r);
            float p_area = (px2 - px1) * (py2 - py1);
            float tx1, ty1, tx2, ty2;
            decode_box(t0, t1, t2, t3, aw, ah, acx, acy, tx1, ty1, tx2, ty2);
            float t_area = (tx2 - tx1) * (ty2 - ty1);
            float xi1 = fmaxf(px1, tx1), yi1 = fmaxf(py1, ty1);
            float xi2 = fminf(px2, tx2), yi2 = fminf(py2, ty2);
            float inter = ((yi2 > yi1) && (xi2 > xi1)) ? (xi2 - xi1) * (yi2 - yi1) : 0.0f;
            float xe1 = fminf(px1, tx1), ye1 = fminf(py1, ty1);
            float xe2 = fmaxf(px2, tx2), ye2 = fmaxf(py2, ty2);
            float enclose = (xe2 - xe1) * (ye2 - ye1) + EPS_F;
            float uni = p_area + t_area - inter + EPS_F;
            float iou_t = inter / uni;
            float giou = iou_t - (enclose - uni) / enclose;
            reg_p = 1.0f - giou;
            float ip = __builtin_nontemporal_load(iou_pred + (size_t)b * A_NUM + a);
            bce_p = fmaxf(ip, 0.0f) - ip * iou_t + log1pf(expf(-fabsf(ip)));
        }
    }

    // ---- single fused WMMA block reduction, deterministic per-block partials --
    float cs, rs, bs, ps;
    block_sum4_256(cls_p, reg_p, bce_p, pos_p, s_red, cs, rs, bs, ps);
    if (tid == 0) {
        float* o = ws_part + (size_t)(blockIdx.y * gridDim.x + blockIdx.x) * 4;
        o[0] = cs;
        o[1] = rs;
        o[2] = bs;
        o[3] = ps;
    }
}

// ---------- kernel 3: deterministic final reduction + scaling ----------------
__global__ void paa_final_kernel(const float* __restrict__ ws_part, int nparts,
                                 float* __restrict__ out) {
    __shared__ float s_red[4][256];
    const int tid = threadIdx.x;
    float cs = 0.0f, rs = 0.0f, bs = 0.0f, ps = 0.0f;
    for (int i = tid; i < nparts; i += 256) {
        const float* p = ws_part + (size_t)i * 4;
        cs += p[0];
        rs += p[1];
        bs += p[2];
        ps += p[3];
    }
    float c2, r2, b2, p2;
    block_sum4_256(cs, rs, bs, ps, s_red, c2, r2, b2, p2);
    if (tid == 0) {
        float np = fmaxf(p2, 1.0f);
        out[0] = c2 / np;
        out[1] = REG_W_F * r2 / np;
        out[2] = IOU_W_F * b2 / np;
    }
}

extern "C" void kernel_launch(void* const* d_in, const int* in_sizes, int n_in,
                              void* d_out, int out_size, void* d_ws, size_t ws_size,
                              hipStream_t stream) {
    const float* box_cls = (const float*)d_in[0];
    const float* box_reg = (const float*)d_in[1];
    const float* iou_pred = (const float*)d_in[2];
    const float* anchors = (const float*)d_in[3];
    const float* gt_boxes = (const float*)d_in[4];
    const int* gt_labels = (const int*)d_in[5];
    float* out = (float*)d_out;

    float* ws_gtbest = (float*)d_ws;            // B*G = 1024 floats
    float* ws_part = ws_gtbest + B_NUM * G_NUM; // nblocks*4 floats

    const int blocks_x = (A_NUM + 255) / 256; // 86
    dim3 grid2(blocks_x, B_NUM);

    paa_gtbest_kernel<<<B_NUM, 256, 0, stream>>>(anchors, gt_boxes, ws_gtbest);
    paa_main_kernel<<<grid2, 256, 0, stream>>>(box_cls, box_reg, iou_pred, anchors,
                                               gt_boxes, gt_labels, ws_gtbest, ws_part);
    paa_final_kernel<<<1, 256, 0, stream>>>(ws_part, blocks_x * B_NUM, out);
}